// FullyConvLCN_13056700580444
// MI455X (gfx1250) — compile-verified
//
#include <hip/hip_runtime.h>

// FullyConvLCN on gfx1250: fused 2-layer locally-connected conv + max,
// layer-1 and layer-2 contractions on V_WMMA_F32_16X16X4_F32.
//
// Shapes: x (16384, 40) f32 ; w1 (64,1,36,5) ; w2 (10,64,32,5) ; out (16384,10)

typedef float v2f __attribute__((ext_vector_type(2)));
typedef float v8f __attribute__((ext_vector_type(8)));

#define WAVES   4
#define BM      (WAVES * 16)   // batch rows per block
#define LIN     40
#define XPAD    48             // padded x row (bank-friendly)
#define C1      64             // layer-1 channels
#define L1OUT   36
#define KW      5
#define C2      10             // layer-2 channels (padded to 16 in N)
#define L2OUT   32
#define SLOT    1024           // floats per ring slot / per-k w2s slab: 32 ch-pairs * 16 * 2(pair)

__device__ __forceinline__ float relu1(float x) {
    // v_med3_f32(x, 0, +inf) == max(x, 0) in one VALU op (finite data)
    return __builtin_amdgcn_fmed3f(x, 0.0f, __builtin_inff());
}
__device__ __forceinline__ float fmax1(float a, float b) {
    // v_med3_f32(a, b, +inf) == max(a, b) in one VALU op (finite data)
    return __builtin_amdgcn_fmed3f(a, b, __builtin_inff());
}

__global__ __launch_bounds__(128, 2)
void lcn_fused_kernel(const float* __restrict__ x,
                      const float* __restrict__ w1,
                      const float* __restrict__ w2,
                      float* __restrict__ out)
{
    // ---- LDS ----
    __shared__ float xs[BM * XPAD];              // 12 KB   staged input rows
    __shared__ float w2s[KW * SLOT];             // 20 KB   w2 slice for current l: [k][c/2][o][c&1]
    __shared__ float ring[WAVES * 5 * SLOT];     // 80 KB   h1 ring: [wave][p%5][c/2][m][c&1]

    const int tid  = threadIdx.x;
    const int wave = tid >> 5;
    const int lane = tid & 31;
    const int nn   = lane & 15;        // A: row m ; B: col n ; D: col n
    const int hi   = lane >> 4;        // 0: K in {0,1} half, 1: K in {2,3} half
    const int d    = hi << 1;          // K sub-offset {0,2}
    const float lom = hi ? 0.0f : 1.0f; // lane mask: k=4 tap lives in K=0 (lanes 0-15) only
    const int b0   = blockIdx.x * BM;  // first batch row of block
    const int b0w  = b0 + wave * 16;   // first batch row of this wave

    float* myring = &ring[wave * 5 * SLOT];
    const int laneAB = hi * 32 + nn * 2;   // per-lane fragment offset (A and B identical)

    // ---- stage x tile: BM x 40 -> LDS (padded to 48) ----
    for (int idx = tid; idx < BM * LIN; idx += 128) {
        int r = idx / LIN, c = idx - r * LIN;
        xs[r * XPAD + c] = x[(b0 + r) * LIN + c];
    }
    __syncthreads();

    // ---- layer 1: compute h1[:, :, p] for block rows 0..15 (xs rows nn, in every wave), ReLU, store to ring ----
    // h1[m, c, p] = relu( sum_k x[m, p+k] * w1[c, p, k] )
    // One 16x16x4 WMMA (k=0..3) + one zero-padded WMMA (k=4) per 16-channel tile.
    auto layer1 = [&](int p) {
        const int slot = p % 5;
        const float* xrow = &xs[nn * XPAD + p];          // A row m = nn
        v2f a;  a.x  = xrow[d];                          // K = d
        a.y  = xrow[d + 1];                              // K = d+1
        v2f a2; a2.x = xrow[4] * lom;                    // tap k=4 in K=0; K=1,2,3 zero
        a2.y = 0.0f;
        #pragma unroll
        for (int ct = 0; ct < 4; ++ct) {
            const int c = ct * 16 + nn;                  // B col n = channel
            const float* wrow = &w1[(c * L1OUT + p) * KW];
            v2f bf;  bf.x  = wrow[d];
            bf.y  = wrow[d + 1];
            v2f bf2; bf2.x = wrow[4] * lom;              // unconditional load, lane-masked by mul
            bf2.y = 0.0f;
            v8f acc = {};
            acc = __builtin_amdgcn_wmma_f32_16x16x4_f32(false, a,  false, bf,  (short)0, acc, false, false);
            acc = __builtin_amdgcn_wmma_f32_16x16x4_f32(false, a2, false, bf2, (short)0, acc, false, false);
            // D: VGPR v, lane -> (m = v + hi*8, channel c). Store ReLU'd pair-interleaved:
            // addr = slot*1024 + (c>>1)*32 + m*2 + (c&1)
            const int base = slot * SLOT + ct * 256 + (nn >> 1) * 32 + (nn & 1) + hi * 16;
            #pragma unroll
            for (int v = 0; v < 8; ++v)
                myring[base + v * 2] = relu1(acc[v]);
        }
    };

    // warm up ring with h1 positions 0..3
    #pragma unroll
    for (int p = 0; p < 4; ++p) layer1(p);

    v8f vmax;
    #pragma unroll
    for (int v = 0; v < 8; ++v) vmax[v] = -3.402823466e+38f;

    // ---- stream output positions l = 0..31 ----
    for (int l = 0; l < L2OUT; ++l) {
        __syncthreads();   // all waves done reading previous w2s slice

        // stage w2 slice for this l into fragment layout: w2s[(k*32 + c/2)*32 + o*2 + (c&1)]
        for (int idx = tid; idx < KW * C1 * 16; idx += 128) {
            const int k = idx % KW;
            const int t = idx / KW;
            const int c = t & 63;
            const int o = t >> 6;          // 0..15 (pad o>=10 with zeros)
            float v = 0.0f;
            if (o < C2) v = w2[(o * C1 + c) * (L2OUT * KW) + l * KW + k];
            w2s[(k * 32 + (c >> 1)) * 32 + o * 2 + (c & 1)] = v;
        }

        // produce h1 position l+4 (wave-private ring, in-order DS within wave)
        layer1(l + 4);

        __syncthreads();   // w2s slice visible to all waves

        // layer 2 at position l: 5 taps x 16 channel-chunks = 80 WMMAs, K-chunks of 4 channels
        v8f acc = {};
        #pragma unroll
        for (int k = 0; k < KW; ++k) {
            const int slot = (l + k) % 5;
            const float* apg = &myring[slot * SLOT + laneAB];
            const float* bpg = &w2s[k * SLOT + laneAB];
            #pragma unroll
            for (int cc = 0; cc < 16; ++cc) {
                v2f af = *(const v2f*)(apg + cc * 64);   // h1[m, c0+d .. c0+d+1, l+k]
                v2f bf = *(const v2f*)(bpg + cc * 64);   // w2[o, c0+d .. c0+d+1, l, k]
                acc = __builtin_amdgcn_wmma_f32_16x16x4_f32(false, af, false, bf, (short)0, acc, false, false);
            }
        }

        #pragma unroll
        for (int v = 0; v < 8; ++v) vmax[v] = fmax1(vmax[v], acc[v]);
    }

    // ---- write out: D layout -> out[b, o], o = nn (<10), m = v + hi*8 ----
    if (nn < C2) {
        #pragma unroll
        for (int v = 0; v < 8; ++v) {
            const int m = v + hi * 8;
            out[(b0w + m) * C2 + nn] = vmax[v];
        }
    }
}

extern "C" void kernel_launch(void* const* d_in, const int* in_sizes, int n_in,
                              void* d_out, int out_size, void* d_ws, size_t ws_size,
                              hipStream_t stream)
{
    const float* x  = (const float*)d_in[0];
    const float* w1 = (const float*)d_in[1];
    const float* w2 = (const float*)d_in[2];
    float* out = (float*)d_out;

    const int B = in_sizes[0] / LIN;       // 16384
    dim3 grid(B / BM), block(128);
    lcn_fused_kernel<<<grid, block, 0, stream>>>(x, w1, w2, out);
}